// EMRouting_39436389712587
// MI455X (gfx1250) — compile-verified
//
#include <hip/hip_runtime.h>
#include <math.h>

// Problem dims (fixed by the reference)
#define B_    8
#define G_    32
#define CIN_  16
#define CL_   3
#define O_    64
#define HW_   1024     // H*W
#define EPS_  1e-7f

#define OT    8        // output-channel tile per workgroup
#define PXT   16       // pixel tile per workgroup (WMMA N)
#define NTHR  128      // 4 wave32 waves
#define LDSST 17       // padded px stride (bank-conflict free)

typedef __attribute__((ext_vector_type(2))) float v2f;
typedef __attribute__((ext_vector_type(8))) float v8f;

__launch_bounds__(NTHR)
__global__ void em_routing_fused(const float* __restrict__ x,
                                 const float* __restrict__ w,
                                 float* __restrict__ out) {
  // con tile: [g][cl][j(0..7)][px(padded 17)]  = 52224 B
  __shared__ float sCon[G_ * CL_ * OT * LDSST];

  const int bid = blockIdx.x;
  const int ot  = bid & 7;          // 8 o-tiles
  const int pt  = (bid >> 3) & 63;  // 64 pixel tiles
  const int b   = bid >> 9;         // batch
  const int o_base = ot * OT;
  const int p_base = pt * PXT;

  const int tid   = threadIdx.x;
  const int lane  = tid & 31;
  const int wv    = tid >> 5;
  const int lo    = lane & 15;      // M row / N col within 16x16 tile
  const int khalf = lane >> 4;      // selects K pair within K=4 step

  // ================= GEMM phase: con = W_g @ x_g via v_wmma_f32_16x16x4_f32 =================
  for (int t = 0; t < G_ / 4; ++t) {
    const int g = wv * (G_ / 4) + t;

    // B matrix (x): B[k][n], n = pixel. lane: n = lo, k-pair = khalf*2 within each K=4 window
    const float* xb = x + (size_t)(b * (G_ * CIN_) + g * CIN_) * HW_ + (p_base + lo);
    v2f breg[4];
#pragma unroll
    for (int kk = 0; kk < 4; ++kk) {
      const int k0 = kk * 4 + khalf * 2;
      breg[kk].x = xb[(size_t)(k0)     * HW_];
      breg[kk].y = xb[(size_t)(k0 + 1) * HW_];
    }

    // A tiles (weight rows). tile0: m=0..15 -> (cl=m/8)*64 + o_base + m%8
    //                        tile1: rows cl=2 (m>=8 duplicated, results discarded)
    const int r0 = (lo >> 3) * O_ + o_base + (lo & 7);
    const int r1 = 2 * O_ + o_base + (lo & 7);
    const float* wr0 = w + (size_t)r0 * (G_ * CIN_) + g * CIN_;
    const float* wr1 = w + (size_t)r1 * (G_ * CIN_) + g * CIN_;

    v8f acc0 = {};
    v8f acc1 = {};
#pragma unroll
    for (int kk = 0; kk < 4; ++kk) {
      const int k0 = kk * 4 + khalf * 2;
      v2f a0; a0.x = wr0[k0]; a0.y = wr0[k0 + 1];
      v2f a1; a1.x = wr1[k0]; a1.y = wr1[k0 + 1];
      acc0 = __builtin_amdgcn_wmma_f32_16x16x4_f32(false, a0, false, breg[kk],
                                                   (short)0, acc0, false, false);
      acc1 = __builtin_amdgcn_wmma_f32_16x16x4_f32(false, a1, false, breg[kk],
                                                   (short)0, acc1, false, false);
    }

    // D layout: VGPR vj holds M = vj + 8*khalf, N = lo
#pragma unroll
    for (int vj = 0; vj < 8; ++vj) {
      const int mm = vj + 8 * khalf;
      const int cl = mm >> 3;          // 0 or 1
      const int j  = mm & 7;
      sCon[((g * CL_ + cl) * OT + j) * LDSST + lo] = acc0[vj];
    }
    if (khalf == 0) {                   // tile1: only M=0..7 are real (cl=2)
#pragma unroll
      for (int vj = 0; vj < 8; ++vj) {
        sCon[((g * CL_ + 2) * OT + vj) * LDSST + lo] = acc1[vj];
      }
    }
  }
  __syncthreads();

  // ================= EM phase: one thread per (o,pixel) site =================
  const int j  = tid >> 4;   // o within tile (0..7)
  const int px = tid & 15;   // pixel within tile

#define CON(gg, cc) sCon[(((gg) * CL_ + (cc)) * OT + j) * LDSST + px]

  const float invG = 1.0f / (float)G_;

  // ---- iter 0: coeff == 1/G ----
  float s1[CL_] = {0.f, 0.f, 0.f}, s2[CL_] = {0.f, 0.f, 0.f};
  for (int g = 0; g < G_; ++g) {
#pragma unroll
    for (int cl = 0; cl < CL_; ++cl) {
      const float c = CON(g, cl);
      s1[cl] += c;
      s2[cl] += c * c;
    }
  }
  float mean[CL_], ivar[CL_], lv[CL_], lpi[CL_];
#pragma unroll
  for (int cl = 0; cl < CL_; ++cl) {
    mean[cl] = s1[cl] * invG;
    const float v = s2[cl] * invG - mean[cl] * mean[cl] + EPS_;
    ivar[cl] = 1.0f / v;
    lv[cl]   = __logf(v);
    lpi[cl]  = __logf(invG);     // pi == 1/G at iter 0
  }

  // ---- iter 1: recompute coeff on the fly from iter-0 stats ----
  float sc[CL_] = {0.f, 0.f, 0.f}, sc2[CL_] = {0.f, 0.f, 0.f}, s0[CL_] = {0.f, 0.f, 0.f};
  for (int g = 0; g < G_; ++g) {
    float c[CL_], tv[CL_];
    float tmax = -INFINITY;
#pragma unroll
    for (int cl = 0; cl < CL_; ++cl) {
      c[cl] = CON(g, cl);
      const float d    = c[cl] - mean[cl];
      const float logp = -0.5f * (d * d * ivar[cl] + lv[cl]);
      tv[cl] = logp * lpi[cl];               // NOTE: multiply, faithful to reference
      tmax = fmaxf(tmax, tv[cl]);
    }
    float e[CL_], es = 0.f;
#pragma unroll
    for (int cl = 0; cl < CL_; ++cl) { e[cl] = __expf(tv[cl] - tmax); es += e[cl]; }
    const float inv = 1.0f / es;
#pragma unroll
    for (int cl = 0; cl < CL_; ++cl) {
      const float co = e[cl] * inv;          // softmax over clusters
      sc[cl]  += co * c[cl];
      sc2[cl] += co * c[cl] * c[cl];
      s0[cl]  += co;
    }
  }
#pragma unroll
  for (int cl = 0; cl < CL_; ++cl) {
    mean[cl] = sc[cl];
    const float v = sc2[cl] - mean[cl] * mean[cl] + EPS_;
    ivar[cl] = 1.0f / v;
    lv[cl]   = __logf(v);
    lpi[cl]  = __logf(s0[cl] * invG);
  }

  // ---- iter 2 (last): mean for cluster 0 only, but softmax needs all clusters ----
  float acc = 0.f;
  for (int g = 0; g < G_; ++g) {
    float c[CL_], tv[CL_];
    float tmax = -INFINITY;
#pragma unroll
    for (int cl = 0; cl < CL_; ++cl) {
      c[cl] = CON(g, cl);
      const float d    = c[cl] - mean[cl];
      const float logp = -0.5f * (d * d * ivar[cl] + lv[cl]);
      tv[cl] = logp * lpi[cl];
      tmax = fmaxf(tmax, tv[cl]);
    }
    const float e0 = __expf(tv[0] - tmax);
    const float e1 = __expf(tv[1] - tmax);
    const float e2 = __expf(tv[2] - tmax);
    acc += (e0 / (e0 + e1 + e2)) * c[0];
  }
#undef CON

  out[(size_t)(b * O_ + o_base + j) * HW_ + (p_base + px)] = acc;
}

extern "C" void kernel_launch(void* const* d_in, const int* in_sizes, int n_in,
                              void* d_out, int out_size, void* d_ws, size_t ws_size,
                              hipStream_t stream) {
  const float* x = (const float*)d_in[0];                 // [B, G*CIN, H, W]
  const float* w = (const float*)d_in[1];                 // [CL*O, G*CIN]
  float* out     = (float*)d_out;                         // [B, O, H, W]
  (void)in_sizes; (void)n_in; (void)out_size; (void)d_ws; (void)ws_size;

  const int nblocks = B_ * (HW_ / PXT) * (O_ / OT);       // 8 * 64 * 8 = 4096
  em_routing_fused<<<nblocks, NTHR, 0, stream>>>(x, w, out);
}